// CDGON_ODE_65532611003017
// MI455X (gfx1250) — compile-verified
//
#include <hip/hip_runtime.h>
#include <hip/hip_bf16.h>

typedef __attribute__((ext_vector_type(2))) float v2f;
typedef __attribute__((ext_vector_type(8))) float v8f;

#define EPSV 1e-4f
#define NND 400
#define DD 64
#define NNE 160000

// ---- fp32 WMMA 16x16x4: D = A(16x4) * B(4x16) + C --------------------------
__device__ __forceinline__ v8f wmma4(v2f a, v2f b, v8f c) {
  return __builtin_amdgcn_wmma_f32_16x16x4_f32(false, a, false, b, (short)0, c,
                                               false, false);
}

// ---- generic  Y[rows x ncols] = act(X [rows x 64]) @ W[64 x ncols] (+bias) --
template <bool DIFF, bool BIAS>
__global__ void gemm_rows64(const float* __restrict__ X, const float* __restrict__ X2,
                            const float* __restrict__ W, const float* __restrict__ bias,
                            float* __restrict__ Y, int rows, int ncols) {
  int wid  = (blockIdx.x * blockDim.x + threadIdx.x) >> 5;
  int lane = threadIdx.x & 31;
  int ntiles = ncols >> 4;
  int rtiles = rows >> 4;
  if (wid >= rtiles * ntiles) return;
  int rt = wid / ntiles, ct = wid - rt * ntiles;
  int m = lane & 15, half = lane >> 4, koff = half * 2;
  int row = rt * 16 + m;
  int n   = ct * 16 + m;
  const float* xr  = X + (size_t)row * DD;
  const float* x2r = DIFF ? (X2 + (size_t)row * DD) : nullptr;
  v8f c = {};
#pragma unroll
  for (int k0 = 0; k0 < DD; k0 += 4) {
    int k = k0 + koff;
    v2f a;
    if constexpr (DIFF) {
      v2f x1 = *(const v2f*)(xr + k);
      v2f x2 = *(const v2f*)(x2r + k);
      a.x = fmaxf(x1.x - x2.x, 0.f);
      a.y = fmaxf(x1.y - x2.y, 0.f);
    } else {
      a = *(const v2f*)(xr + k);
    }
    v2f b; b.x = W[k * ncols + n]; b.y = W[(k + 1) * ncols + n];
    c = wmma4(a, b, c);
  }
#pragma unroll
  for (int r = 0; r < 8; ++r) {
    int ro = rt * 16 + r + half * 8;
    float v = c[r];
    if constexpr (BIAS) v += bias[n];
    Y[(size_t)ro * ncols + n] = v;
  }
}

// ---- per-row scalar MLP2: out[r] = relu(x@W1+b1)@W2 + b2 -------------------
__global__ void mlp2_rows(const float* __restrict__ X, const float* __restrict__ W1,
                          const float* __restrict__ b1, const float* __restrict__ W2,
                          const float* __restrict__ b2, float* __restrict__ out) {
  int r = blockIdx.x * blockDim.x + threadIdx.x;
  if (r >= NND) return;
  float h[32];
#pragma unroll
  for (int j = 0; j < 32; ++j) h[j] = b1[j];
  for (int k = 0; k < 64; ++k) {
    float xv = X[(size_t)r * 64 + k];
#pragma unroll
    for (int j = 0; j < 32; ++j) h[j] += xv * W1[k * 32 + j];
  }
  float acc = b2[0];
#pragma unroll
  for (int j = 0; j < 32; ++j) acc += fmaxf(h[j], 0.f) * W2[j];
  out[r] = acc;
}

// ---- A[i,j] = relu( sum_k relu(PA[i,k]+QA[j,k]) * Wev2[k] + bev2 ) ---------
__global__ void compute_A(const float* __restrict__ PA, const float* __restrict__ QA,
                          const float* __restrict__ Wev2, const float* __restrict__ bev2,
                          float* __restrict__ Aout) {
  __shared__ float Ps[16][33], Qs[16][33];
  int bi = blockIdx.x / 25, bj = blockIdx.x % 25;
  for (int l = threadIdx.x; l < 512; l += 256) {
    int r = l >> 5, k = l & 31;
    Ps[r][k] = PA[(size_t)(bi * 16 + r) * 32 + k];
    Qs[r][k] = QA[(size_t)(bj * 16 + r) * 32 + k];
  }
  __syncthreads();
  int ti = threadIdx.x >> 4, tj = threadIdx.x & 15;
  float acc = bev2[0];
#pragma unroll
  for (int k = 0; k < 32; ++k) acc += fmaxf(Ps[ti][k] + Qs[tj][k], 0.f) * Wev2[k];
  Aout[(size_t)(bi * 16 + ti) * NND + bj * 16 + tj] = fmaxf(acc, 0.f);
}

// ---- dh[i] = rsqrt(max(rowsum(A),eps)) -------------------------------------
__global__ void rowsum_dh(const float* __restrict__ Aws, float* __restrict__ dh) {
  __shared__ float sd[128];
  int i = blockIdx.x;
  float s = 0.f;
  for (int j = threadIdx.x; j < NND; j += 128) s += Aws[(size_t)i * NND + j];
  sd[threadIdx.x] = s;
  __syncthreads();
  for (int o = 64; o > 0; o >>= 1) {
    if (threadIdx.x < o) sd[threadIdx.x] += sd[threadIdx.x + o];
    __syncthreads();
  }
  if (threadIdx.x == 0) dh[i] = rsqrtf(fmaxf(sd[0], EPSV));
}

// ---- massage = (d A d) @ NW3  (400x400x64, scale applied on the fly) -------
__global__ void gemm_normA(const float* __restrict__ Aws, const float* __restrict__ dh,
                           const float* __restrict__ NW3, float* __restrict__ Y) {
  int wid  = (blockIdx.x * blockDim.x + threadIdx.x) >> 5;
  int lane = threadIdx.x & 31;
  if (wid >= 100) return;
  int rt = wid >> 2, ct = wid & 3;
  int m = lane & 15, half = lane >> 4, koff = half * 2;
  int row = rt * 16 + m;
  float dr = dh[row];
  const float* Ar = Aws + (size_t)row * NND;
  int n = ct * 16 + m;
  v8f c = {};
  for (int k0 = 0; k0 < NND; k0 += 4) {
    int k = k0 + koff;
    v2f av = *(const v2f*)(Ar + k);
    v2f dp = *(const v2f*)(dh + k);
    v2f a; a.x = dr * av.x * dp.x; a.y = dr * av.y * dp.y;
    v2f b; b.x = NW3[(size_t)k * 64 + n]; b.y = NW3[(size_t)(k + 1) * 64 + n];
    c = wmma4(a, b, c);
  }
#pragma unroll
  for (int r = 0; r < 8; ++r)
    Y[(size_t)(rt * 16 + r + half * 8) * 64 + n] = c[r];
}

__global__ void init_mab(unsigned* mab) { if (threadIdx.x == 0) mab[0] = 0u; }

__global__ void maxabs_kernel(const float* __restrict__ alpha, unsigned* __restrict__ mab) {
  int tid = blockIdx.x * blockDim.x + threadIdx.x;
  if (tid < NND * 64) atomicMax(mab, __float_as_uint(fabsf(alpha[tid])));
}

// ---- edge kernel: G=E@Wg, F=E@We (WMMA), softmax(G), write M/pred_edge/ind -
#define LP 65
__global__ void edge_kernel(const float* __restrict__ edge_emb,
                            const float* __restrict__ Wg, const float* __restrict__ bg,
                            const float* __restrict__ We, const float* __restrict__ be,
                            const float* __restrict__ P2, const float* __restrict__ Q2,
                            const float* __restrict__ Aws,
                            float* __restrict__ M, float* __restrict__ out) {
  __shared__ float Gs[4][16 * LP];
  __shared__ float Fs[4][16 * LP];
  __shared__ float rmx[4][16];
  __shared__ float rsm[4][16];
  int w = threadIdx.x >> 5, lane = threadIdx.x & 31;
  int t  = blockIdx.x * 4 + w;           // 10000 tiles exactly
  int e0 = t * 16;
  int m = lane & 15, half = lane >> 4, koff = half * 2;
  const float* E = edge_emb + (size_t)(e0 + m) * 64;
  v8f cg[4] = {{}, {}, {}, {}};
  v8f cf[4] = {{}, {}, {}, {}};
#pragma unroll 4
  for (int k0 = 0; k0 < 64; k0 += 4) {
    int k = k0 + koff;
    v2f a = *(const v2f*)(E + k);
#pragma unroll
    for (int ct = 0; ct < 4; ++ct) {
      int n = ct * 16 + m;
      v2f bgv; bgv.x = Wg[k * 64 + n]; bgv.y = Wg[(k + 1) * 64 + n];
      cg[ct] = wmma4(a, bgv, cg[ct]);
      v2f bev; bev.x = We[k * 64 + n]; bev.y = We[(k + 1) * 64 + n];
      cf[ct] = wmma4(a, bev, cf[ct]);
    }
  }
#pragma unroll
  for (int ct = 0; ct < 4; ++ct) {
    int col = ct * 16 + m;
#pragma unroll
    for (int r = 0; r < 8; ++r) {
      int row = r + half * 8;
      Gs[w][row * LP + col] = cg[ct][r] + bg[col];
      Fs[w][row * LP + col] = cf[ct][r] + be[col];
    }
  }
  __syncthreads();
  if (lane < 16) {
    float mx = -1e30f;
    for (int c = 0; c < 64; ++c) mx = fmaxf(mx, Gs[w][lane * LP + c]);
    float s = 0.f;
    for (int c = 0; c < 64; ++c) s += expf(Gs[w][lane * LP + c] - mx);
    rmx[w][lane] = mx;
    rsm[w][lane] = s;
  }
  __syncthreads();
  for (int idx = lane; idx < 1024; idx += 32) {
    int r = idx >> 6, c = idx & 63;
    int e = e0 + r;
    float av  = Aws[e];
    float ind = (av < 1.0f) ? 0.f : 1.f;
    int i = e / NND;
    int j = e - i * NND;
    float g  = expf(Gs[w][r * LP + c] - rmx[w][r]) / rsm[w][r];
    float ev = edge_emb[(size_t)e * 64 + c];
    M[(size_t)e * 64 + c] = ind * g * ev;
    float pe = ind * (Fs[w][r * LP + c] + P2[(size_t)i * 64 + c] + Q2[(size_t)j * 64 + c]);
    out[(size_t)(NND + e) * 64 + c] = pe;
    if (c == 0) out[(size_t)160400 * 64 + e] = ind;  // edge_indicator output
  }
}

// ---- split-K  partial[ch] = A_ne(16x800 chunk) @ M(800x64) -----------------
#define NCH 200
#define KCH 800
__global__ void aggr_kernel(const float* __restrict__ A_ne, const float* __restrict__ M,
                            float* __restrict__ part) {
  int wid  = (blockIdx.x * blockDim.x + threadIdx.x) >> 5;
  int lane = threadIdx.x & 31;
  int rt = wid / NCH, ch = wid - rt * NCH;
  if (rt >= 25) return;
  int m = lane & 15, half = lane >> 4, koff = half * 2;
  const float* Ar = A_ne + (size_t)(rt * 16 + m) * NNE + ch * KCH + koff;
  const float* Mr = M + (size_t)(ch * KCH + koff) * 64;
  v8f c0 = {}, c1 = {}, c2 = {}, c3 = {};
#pragma unroll 2
  for (int kk = 0; kk < KCH; kk += 4) {
    v2f a = *(const v2f*)(Ar);
    Ar += 4;
    v2f b;
    b.x = Mr[m];      b.y = Mr[64 + m];      c0 = wmma4(a, b, c0);
    b.x = Mr[16 + m]; b.y = Mr[64 + 16 + m]; c1 = wmma4(a, b, c1);
    b.x = Mr[32 + m]; b.y = Mr[64 + 32 + m]; c2 = wmma4(a, b, c2);
    b.x = Mr[48 + m]; b.y = Mr[64 + 48 + m]; c3 = wmma4(a, b, c3);
    Mr += 4 * 64;
  }
  float* P = part + (size_t)ch * (NND * 64);
#pragma unroll
  for (int r = 0; r < 8; ++r) {
    int ro = rt * 16 + r + half * 8;
    P[(size_t)ro * 64 + m]      = c0[r];
    P[(size_t)ro * 64 + 16 + m] = c1[r];
    P[(size_t)ro * 64 + 32 + m] = c2[r];
    P[(size_t)ro * 64 + 48 + m] = c3[r];
  }
}

// ---- reduce split-K partials + assemble pred_node --------------------------
__global__ void finalize_node(const float* __restrict__ part, const float* __restrict__ alpha,
                              const unsigned* __restrict__ mab, const float* __restrict__ numer,
                              const float* __restrict__ denom, const float* __restrict__ RW4,
                              float* __restrict__ out) {
  int tid = blockIdx.x * blockDim.x + threadIdx.x;
  if (tid >= NND * 64) return;
  float s = 0.f;
  for (int ch = 0; ch < NCH; ++ch) s += part[(size_t)ch * (NND * 64) + tid];
  int i = tid >> 6;
  float amax = fmaxf(__uint_as_float(mab[0]), EPSV);
  float al   = alpha[tid] / amax;
  float dn   = denom[i];
  dn = (dn == 0.f) ? EPSV : dn;
  float nv = numer[i] / dn;
  out[tid] = al * (nv * RW4[tid]) + s;
}

extern "C" void kernel_launch(void* const* d_in, const int* in_sizes, int n_in,
                              void* d_out, int out_size, void* d_ws, size_t ws_size,
                              hipStream_t stream) {
  const float* nee     = (const float*)d_in[1];
  const float* flow    = (const float*)d_in[2];
  const float* A_ne    = (const float*)d_in[3];
  const float* W_alpha = (const float*)d_in[5];
  const float* b_alpha = (const float*)d_in[6];
  const float* W_gamma = (const float*)d_in[7];
  const float* b_gamma = (const float*)d_in[8];
  const float* W_n2e   = (const float*)d_in[9];
  const float* b_n2e   = (const float*)d_in[10];
  const float* W_ev1   = (const float*)d_in[11];
  const float* b_ev1   = (const float*)d_in[12];
  const float* W_ev2   = (const float*)d_in[13];
  const float* b_ev2   = (const float*)d_in[14];
  const float* W_ft1   = (const float*)d_in[15];
  const float* b_ft1   = (const float*)d_in[16];
  const float* W_ft2   = (const float*)d_in[17];
  const float* b_ft2   = (const float*)d_in[18];
  const float* W3      = (const float*)d_in[19];
  const float* W4      = (const float*)d_in[20];
  const float* W_e2e   = (const float*)d_in[21];
  const float* b_e2e   = (const float*)d_in[22];
  const float* W_n2e2  = (const float*)d_in[23];
  const float* b_n2e2  = (const float*)d_in[24];

  const float* node_emb = nee;
  const float* edge_emb = nee + (size_t)NND * 64;

  float* ws      = (float*)d_ws;
  float* Pn      = ws;                    // 25600  (includes +b_n2e)
  float* Qn      = Pn + 25600;            // 25600
  float* P2      = Qn + 25600;            // 25600  (includes +b_n2e2)
  float* Q2      = P2 + 25600;            // 25600
  float* NW3     = Q2 + 25600;            // 25600
  float* RW4     = NW3 + 25600;           // 25600
  float* PA      = RW4 + 25600;           // 12800  (includes +b_ev1)
  float* QA      = PA + 12800;            // 12800
  float* Aws     = QA + 12800;            // 160000
  float* dh      = Aws + 160000;          // 400
  float* numer   = dh + 400;              // 400
  float* denom   = numer + 400;           // 400
  float* massage = denom + 400;           // 25600
  float* alpha   = massage + 25600;       // 25600
  unsigned* mab  = (unsigned*)(alpha + 25600);
  float* M       = alpha + 25600 + 64;    // 160000*64
  float* part    = M + (size_t)NNE * 64;  // 200*25600

  float* out = (float*)d_out;

  init_mab<<<1, 32, 0, stream>>>(mab);

  gemm_rows64<false, true ><<<25, 128, 0, stream>>>(node_emb, nullptr, W_n2e,         b_n2e,   Pn,  NND, 64);
  gemm_rows64<false, false><<<25, 128, 0, stream>>>(node_emb, nullptr, W_n2e + 4096,  nullptr, Qn,  NND, 64);
  gemm_rows64<false, true ><<<25, 128, 0, stream>>>(node_emb, nullptr, W_n2e2,        b_n2e2,  P2,  NND, 64);
  gemm_rows64<false, false><<<25, 128, 0, stream>>>(node_emb, nullptr, W_n2e2 + 4096, nullptr, Q2,  NND, 64);
  gemm_rows64<false, false><<<25, 128, 0, stream>>>(node_emb, nullptr, W3,            nullptr, NW3, NND, 64);
  gemm_rows64<true,  false><<<25, 128, 0, stream>>>(flow,     node_emb, W4,           nullptr, RW4, NND, 64);
  gemm_rows64<false, true ><<<13, 128, 0, stream>>>(Pn, nullptr, W_ev1, b_ev1,   PA, NND, 32);
  gemm_rows64<false, false><<<13, 128, 0, stream>>>(Qn, nullptr, W_ev1, nullptr, QA, NND, 32);

  mlp2_rows<<<2, 256, 0, stream>>>(node_emb, W_ft1, b_ft1, W_ft2, b_ft2, numer);
  mlp2_rows<<<2, 256, 0, stream>>>(flow,     W_ft1, b_ft1, W_ft2, b_ft2, denom);

  compute_A<<<625, 256, 0, stream>>>(PA, QA, W_ev2, b_ev2, Aws);
  rowsum_dh<<<400, 128, 0, stream>>>(Aws, dh);
  gemm_normA<<<25, 128, 0, stream>>>(Aws, dh, NW3, massage);
  gemm_rows64<false, true><<<25, 128, 0, stream>>>(massage, nullptr, W_alpha, b_alpha, alpha, NND, 64);
  maxabs_kernel<<<100, 256, 0, stream>>>(alpha, mab);

  edge_kernel<<<2500, 128, 0, stream>>>(edge_emb, W_gamma, b_gamma, W_e2e, b_e2e,
                                        P2, Q2, Aws, M, out);
  aggr_kernel<<<1250, 128, 0, stream>>>(A_ne, M, part);
  finalize_node<<<100, 256, 0, stream>>>(part, alpha, mab, numer, denom, RW4, out);

  (void)in_sizes; (void)n_in; (void)out_size; (void)ws_size;
}